// InformedSender_87900800680732
// MI455X (gfx1250) — compile-verified
//
#include <hip/hip_runtime.h>
#include <math.h>

typedef __attribute__((ext_vector_type(2))) float v2f;
typedef __attribute__((ext_vector_type(8))) float v8f;

#define G_ 2
#define N_ 20000
#define E_ 320000
#define F_ 4096
#define C_ 64          /* hid == emb == 64 */
#define VOCAB_ 10
#define ATT_SLOPE 0.2f
#define IM_SLOPE 0.01f

__device__ __forceinline__ float leakyf(float x, float s) { return x >= 0.f ? x : s * x; }

/* monotonic float -> uint encoding so atomicMax(uint) == float max; 0 encodes "-inf" sentinel */
__device__ __forceinline__ unsigned fenc(float x) {
    unsigned u = __float_as_uint(x);
    return (u & 0x80000000u) ? ~u : (u | 0x80000000u);
}
__device__ __forceinline__ float fdec(unsigned u) {
    return (u & 0x80000000u) ? __uint_as_float(u ^ 0x80000000u) : __uint_as_float(~u);
}

__global__ void k_fill(float* p, float v, int n) {
    int t = blockIdx.x * blockDim.x + threadIdx.x;
    if (t < n) p[t] = v;
}

/* h[n,c] = node_feat[n,:3] @ gat1_w[:, c]  (head 0 => cols 0..63 of [3,128]) */
__global__ void k_node_h1(const float* __restrict__ nf, const float* __restrict__ w,
                          float* __restrict__ h) {
    int t = blockIdx.x * blockDim.x + threadIdx.x;
    if (t >= N_ * C_) return;
    int n = t >> 6, c = t & 63;
    const float* f = nf + n * 3;
    h[t] = f[0] * w[c] + f[1] * w[128 + c] + f[2] * w[256 + c];
}

/* per-node attention logits: el[n]=h[n,:].al, er[n]=h[n,:].ar — one wave32 per node */
__global__ void k_node_elr(const float* __restrict__ h, const float* __restrict__ al,
                           const float* __restrict__ ar, float* __restrict__ el,
                           float* __restrict__ er) {
    int wave = (blockIdx.x * blockDim.x + threadIdx.x) >> 5;
    int lane = threadIdx.x & 31;
    if (wave >= N_) return;
    const float* hr = h + wave * 64;
    float a = hr[lane] * al[lane] + hr[lane + 32] * al[lane + 32];
    float b = hr[lane] * ar[lane] + hr[lane + 32] * ar[lane + 32];
    for (int off = 16; off > 0; off >>= 1) {
        a += __shfl_xor(a, off, 32);
        b += __shfl_xor(b, off, 32);
    }
    if (lane == 0) { el[wave] = a; er[wave] = b; }
}

__global__ void k_edge_max(const int* __restrict__ src, const int* __restrict__ dst,
                           const float* __restrict__ el, const float* __restrict__ er,
                           unsigned* __restrict__ menc) {
    int t = blockIdx.x * blockDim.x + threadIdx.x;
    if (t >= E_) return;
    int d = dst[t];
    float v = leakyf(el[src[t]] + er[d], ATT_SLOPE);
    atomicMax(&menc[d], fenc(v));
}

__global__ void k_edge_expsum(const int* __restrict__ src, const int* __restrict__ dst,
                              const float* __restrict__ el, const float* __restrict__ er,
                              const unsigned* __restrict__ menc, float* __restrict__ ee,
                              float* __restrict__ den) {
    int t = blockIdx.x * blockDim.x + threadIdx.x;
    if (t >= E_) return;
    int d = dst[t];
    float v = leakyf(el[src[t]] + er[d], ATT_SLOPE);
    float e = expf(v - fdec(menc[d]));
    ee[t] = e;
    atomicAdd(&den[d], e);
}

/* msg scatter: agg[dst, c] += h[src, c] * (ee/den[dst]); one thread per (edge, channel) */
__global__ void k_edge_agg(const int* __restrict__ src, const int* __restrict__ dst,
                           const float* __restrict__ ee, const float* __restrict__ den,
                           const float* __restrict__ h, float* __restrict__ agg) {
    int t = blockIdx.x * blockDim.x + threadIdx.x;
    if (t >= E_ * 64) return;
    int e = t >> 6, c = t & 63;
    int d = dst[e];
    float a = ee[e] / den[d];
    atomicAdd(&agg[d * 64 + c], h[src[e] * 64 + c] * a);
}

__global__ void k_relu_bias(float* __restrict__ agg, const float* __restrict__ b) {
    int t = blockIdx.x * blockDim.x + threadIdx.x;
    if (t >= N_ * C_) return;
    float v = agg[t] + b[t & 63];
    agg[t] = v > 0.f ? v : 0.f;
}

/* D[20000,64] = A[20000,64] @ W[:,0:64] (W row-major [64,128]) via V_WMMA_F32_16X16X4_F32.
 * One wave per 16x16 output tile. Layouts per ISA 7.12.2:
 *  A 16x4: lane<16 holds row M=lane, {K, K+1}; lane>=16 holds {K+2, K+3}
 *  B 4x16: mirrors A's K split across lane halves; N = lane&15
 *  C/D:    VGPR v -> M = v (+8 for upper lanes), N = lane&15                    */
__global__ void k_gemm_wmma(const float* __restrict__ A, const float* __restrict__ W,
                            float* __restrict__ D) {
    int tm = blockIdx.x;              /* 0..1249 */
    int tn = blockIdx.y;              /* 0..3    */
    int lane = threadIdx.x & 31;
    int half = lane >> 4, lrow = lane & 15;
    const float* Arow = A + (tm * 16 + lrow) * 64;
    int ncol = tn * 16 + lrow;
    v8f c = {};
#pragma unroll
    for (int k = 0; k < 64; k += 4) {
        v2f a, b;
        a.x = Arow[k + 2 * half];
        a.y = Arow[k + 2 * half + 1];
        b.x = W[(k + 2 * half) * 128 + ncol];
        b.y = W[(k + 2 * half + 1) * 128 + ncol];
        c = __builtin_amdgcn_wmma_f32_16x16x4_f32(false, a, false, b, (short)0, c, false, false);
    }
#pragma unroll
    for (int v = 0; v < 8; v++) {
        int m = tm * 16 + v + 8 * half;
        D[m * 64 + ncol] = c[v];
    }
}

/* gr[c] = mean_n agg[n,c] + b2[c] */
__global__ void k_colmean(const float* __restrict__ agg, const float* __restrict__ b2,
                          float* __restrict__ gr) {
    int c = blockIdx.x, t = threadIdx.x;
    __shared__ float s[256];
    float acc = 0.f;
    for (int n = t; n < N_; n += 256) acc += agg[n * 64 + c];
    s[t] = acc; __syncthreads();
    for (int o = 128; o > 0; o >>= 1) { if (t < o) s[t] += s[t + o]; __syncthreads(); }
    if (t == 0) gr[c] = s[0] / (float)N_ + b2[c];
}

/* image path: E[g,j] = x[g,:] @ im_lin1_w[:,j] ; one block per (g,j) */
__global__ void k_im_embed(const float* __restrict__ x, const float* __restrict__ w,
                           float* __restrict__ Ef) {
    int g = blockIdx.x >> 6, j = blockIdx.x & 63, t = threadIdx.x;
    __shared__ float s[256];
    float acc = 0.f;
    for (int k = t; k < F_; k += 256) acc += x[g * F_ + k] * w[k * 64 + j];
    s[t] = acc; __syncthreads();
    for (int o = 128; o > 0; o >>= 1) { if (t < o) s[t] += s[t + o]; __syncthreads(); }
    if (t == 0) Ef[g * 64 + j] = s[0];
}

/* h1 = leaky(conv1_w @ E); h2 = leaky(conv2_w @ h1) */
__global__ void k_im_head(const float* __restrict__ Ef, const float* __restrict__ c1,
                          const float* __restrict__ c2, float* __restrict__ h2) {
    __shared__ float h1[64 * 64];
    for (int idx = threadIdx.x; idx < 4096; idx += blockDim.x) {
        int i = idx >> 6, j = idx & 63;
        h1[idx] = leakyf(c1[i * 2 + 0] * Ef[j] + c1[i * 2 + 1] * Ef[64 + j], IM_SLOPE);
    }
    __syncthreads();
    if (threadIdx.x < 64) {
        int j = threadIdx.x;
        float s = 0.f;
        for (int i = 0; i < 64; i++) s += c2[i] * h1[i * 64 + j];
        h2[j] = leakyf(s, IM_SLOPE);
    }
}

/* grl = gr@W + b; h = [grl, h2]; out = log_softmax(h @ im_lin2_w) */
__global__ void k_final(const float* __restrict__ gr, const float* __restrict__ h2,
                        const float* __restrict__ w1, const float* __restrict__ b1,
                        const float* __restrict__ w2, float* __restrict__ out) {
    __shared__ float hcat[128];
    __shared__ float logits[VOCAB_];
    __shared__ float lse;
    int t = threadIdx.x;
    if (t < 64) {
        float s = b1[t];
        for (int i = 0; i < 128; i++) s += gr[i] * w1[i * 64 + t];
        hcat[t] = s;
    } else {
        hcat[t] = h2[t - 64];
    }
    __syncthreads();
    if (t < VOCAB_) {
        float s = 0.f;
        for (int i = 0; i < 128; i++) s += hcat[i] * w2[i * VOCAB_ + t];
        logits[t] = s;
    }
    __syncthreads();
    if (t == 0) {
        float m = logits[0];
        for (int i = 1; i < VOCAB_; i++) m = fmaxf(m, logits[i]);
        float s = 0.f;
        for (int i = 0; i < VOCAB_; i++) s += expf(logits[i] - m);
        lse = m + logf(s);
    }
    __syncthreads();
    if (t < VOCAB_) out[t] = logits[t] - lse;
}

extern "C" void kernel_launch(void* const* d_in, const int* in_sizes, int n_in,
                              void* d_out, int out_size, void* d_ws, size_t ws_size,
                              hipStream_t stream) {
    (void)in_sizes; (void)n_in; (void)out_size; (void)ws_size;
    const float* x       = (const float*)d_in[0];
    const float* nodef   = (const float*)d_in[1];
    const int*   src     = (const int*)d_in[2];
    const int*   dst     = (const int*)d_in[3];
    const float* im1w    = (const float*)d_in[4];
    const float* conv1w  = (const float*)d_in[5];
    const float* conv2w  = (const float*)d_in[6];
    const float* g1w     = (const float*)d_in[7];
    const float* g1al    = (const float*)d_in[8];   /* head 0 = first 64 */
    const float* g1ar    = (const float*)d_in[9];
    const float* g1b     = (const float*)d_in[10];
    const float* g2w     = (const float*)d_in[11];
    const float* g2al    = (const float*)d_in[12];
    const float* g2ar    = (const float*)d_in[13];
    const float* g2b     = (const float*)d_in[14];
    const float* grw     = (const float*)d_in[15];
    const float* grb     = (const float*)d_in[16];
    const float* im2w    = (const float*)d_in[17];
    float* out = (float*)d_out;

    float* ws = (float*)d_ws;
    size_t o = 0;
    float*    bufA = ws + o; o += (size_t)N_ * C_;
    float*    bufB = ws + o; o += (size_t)N_ * C_;
    float*    el   = ws + o; o += N_;
    float*    er   = ws + o; o += N_;
    unsigned* menc = (unsigned*)(ws + o); o += N_;
    float*    den  = ws + o; o += N_;
    float*    ee   = ws + o; o += E_;
    float*    Ef   = ws + o; o += 128;
    float*    h2   = ws + o; o += 64;
    float*    gr   = ws + o; o += 128;

    const int NC = N_ * C_;
    const int bNC = (NC + 255) / 256, bE = (E_ + 255) / 256;
    const int bEC = (E_ * 64 + 255) / 256, bN = (N_ + 255) / 256;

    /* image path (independent) */
    k_im_embed<<<128, 256, 0, stream>>>(x, im1w, Ef);
    k_im_head<<<1, 256, 0, stream>>>(Ef, conv1w, conv2w, h2);

    for (int g = 0; g < G_; g++) {
        const float* nf = nodef + (size_t)g * N_ * 3;
        const int* sg = src + (size_t)g * E_;
        const int* dg = dst + (size_t)g * E_;

        /* ---- GAT layer 1, head 0 ---- */
        k_node_h1<<<bNC, 256, 0, stream>>>(nf, g1w, bufA);
        k_node_elr<<<(N_ + 7) / 8, 256, 0, stream>>>(bufA, g1al, g1ar, el, er);
        k_fill<<<bN, 256, 0, stream>>>((float*)menc, 0.f, N_);
        k_fill<<<bN, 256, 0, stream>>>(den, 0.f, N_);
        k_fill<<<bNC, 256, 0, stream>>>(bufB, 0.f, NC);
        k_edge_max<<<bE, 256, 0, stream>>>(sg, dg, el, er, menc);
        k_edge_expsum<<<bE, 256, 0, stream>>>(sg, dg, el, er, menc, ee, den);
        k_edge_agg<<<bEC, 256, 0, stream>>>(sg, dg, ee, den, bufA, bufB);
        k_relu_bias<<<bNC, 256, 0, stream>>>(bufB, g1b);   /* bufB = r1 (head0), relu'd */

        /* ---- GAT layer 2, head 0 ---- */
        k_gemm_wmma<<<dim3(N_ / 16, 4), 32, 0, stream>>>(bufB, g2w, bufA);  /* bufA = h2g */
        k_node_elr<<<(N_ + 7) / 8, 256, 0, stream>>>(bufA, g2al, g2ar, el, er);
        k_fill<<<bN, 256, 0, stream>>>((float*)menc, 0.f, N_);
        k_fill<<<bN, 256, 0, stream>>>(den, 0.f, N_);
        k_fill<<<bNC, 256, 0, stream>>>(bufB, 0.f, NC);
        k_edge_max<<<bE, 256, 0, stream>>>(sg, dg, el, er, menc);
        k_edge_expsum<<<bE, 256, 0, stream>>>(sg, dg, el, er, menc, ee, den);
        k_edge_agg<<<bEC, 256, 0, stream>>>(sg, dg, ee, den, bufA, bufB);
        k_colmean<<<64, 256, 0, stream>>>(bufB, g2b, gr + g * 64);
    }

    k_final<<<1, 128, 0, stream>>>(gr, h2, grw, grb, im2w, out);
}